// RWLoss_79242146611971
// MI455X (gfx1250) — compile-verified
//
#include <hip/hip_runtime.h>

// Problem dims fixed by setup_inputs(): x[B,C,H,W] f32, y_[B,1,H,W] int.
constexpr int DB = 2;
constexpr int DC = 4;
constexpr int DH = 256;
constexpr int DW = 256;
constexpr int HWSZ = DH * DW;
constexpr float BIGD = (float)(DH + DW);   // matches reference BIG = H + W

typedef __attribute__((ext_vector_type(2))) float v2f;
typedef __attribute__((ext_vector_type(8))) float v8f;

// ---------------------------------------------------------------------------
// Kernel 1: per-column forward/backward scan -> vertical distance g, store g^2.
// One thread per (b,c,w) column: B*C*W = 2048 threads. Also zero-init maxd2.
// ---------------------------------------------------------------------------
__global__ void edt_cols_kernel(const int* __restrict__ y,
                                float* __restrict__ field,
                                int* __restrict__ maxd2) {
    const int q = blockIdx.x * blockDim.x + threadIdx.x;   // [0, B*C*W)
    if (blockIdx.x == 0 && threadIdx.x < DB * DC) maxd2[threadIdx.x] = 0; // 0 == bits of 0.0f
    if (q >= DB * DC * DW) return;

    const int w  = q % DW;
    const int bc = q / DW;
    const int b  = bc / DC;
    const int c  = bc % DC;

    const int* lab = y + b * HWSZ + w;          // stride DW along h
    float*     col = field + bc * HWSZ + w;     // stride DW along h

    // forward pass: df[h] = steps since last (label != c), init BIG
    float df = BIGD;
    for (int h = 0; h < DH; ++h) {
        const bool m = (lab[h * DW] != c);
        df = m ? 0.0f : (df + 1.0f);
        col[h * DW] = df;
    }
    // backward pass, combine, square
    float db = BIGD;
    for (int h = DH - 1; h >= 0; --h) {
        const bool m = (lab[h * DW] != c);
        db = m ? 0.0f : (db + 1.0f);
        const float g = fminf(col[h * DW], db);
        col[h * DW] = g * g;                    // exact: g <= 768 -> g^2 < 2^24
    }
}

// ---------------------------------------------------------------------------
// Kernel 2: per-row min-plus: d2[w] = min_{w'} (g2[w'] + (w-w')^2), in place.
// One block per (b,c,h) row: 2048 blocks x 256 threads. Block-max -> atomicMax.
// All intermediates are exact fp32 integers -> bit-identical to the reference.
// ---------------------------------------------------------------------------
__global__ void edt_rows_kernel(float* __restrict__ field,
                                int* __restrict__ maxd2) {
    __shared__ float s_row[DW];
    __shared__ float s_red[256];

    const int row = blockIdx.x;                 // [0, B*C*H)
    const int w   = threadIdx.x;
    float* rp = field + row * DW;

    s_row[w] = rp[w];
    __syncthreads();

    const float fw = (float)w;
    float best = 3.0e38f;
    for (int wp = 0; wp < DW; ++wp) {           // broadcast LDS reads (same addr all lanes)
        const float dx = fw - (float)wp;
        best = fminf(best, fmaf(dx, dx, s_row[wp]));
    }
    rp[w] = best;                               // overwrite g2 with d2 (same-row dependency only)

    // deterministic block max of d2
    s_red[w] = best;
    __syncthreads();
    for (int s = 128; s > 0; s >>= 1) {
        if (w < s) s_red[w] = fmaxf(s_red[w], s_red[w + s]);
        __syncthreads();
    }
    if (w == 0) {
        // d2 >= 0, so int compare == float compare on the raw bits (deterministic)
        atomicMax(&maxd2[row / DH], __float_as_int(s_red[0]));
    }
}

// ---------------------------------------------------------------------------
// Kernel 3: per-pixel softmax over C, rrw weighting, block partial sums.
// 512 blocks x 256 threads over B*H*W pixels.
// ---------------------------------------------------------------------------
__global__ void loss_partial_kernel(const float* __restrict__ x,
                                    const float* __restrict__ d2f,
                                    const int* __restrict__ maxd2,
                                    float* __restrict__ partial) {
    __shared__ float s_red[256];

    const int p  = blockIdx.x * 256 + threadIdx.x;   // [0, B*H*W)
    const int b  = p / HWSZ;
    const int hw = p % HWSZ;

    float logit[DC];
    float mx = -3.0e38f;
#pragma unroll
    for (int c = 0; c < DC; ++c) {
        logit[c] = x[(b * DC + c) * HWSZ + hw];
        mx = fmaxf(mx, logit[c]);
    }
    float e[DC];
    float se = 0.0f;
#pragma unroll
    for (int c = 0; c < DC; ++c) {
        e[c] = expf(logit[c] - mx);
        se += e[c];
    }
    const float inv_se = 1.0f / se;

    float t = 0.0f;
#pragma unroll
    for (int c = 0; c < DC; ++c) {
        const float d2 = d2f[(b * DC + c) * HWSZ + hw];
        float r;
        if (d2 == 0.0f) {
            r = 1.0f;                                   // where(rrw==0, 1, rrw)
        } else {
            const float maxd = sqrtf(__int_as_float(maxd2[b * DC + c]));
            r = -(sqrtf(d2) / (maxd + 1e-15f));
        }
        t += (e[c] * inv_se) * r;
    }

    // deterministic block tree-sum
    s_red[threadIdx.x] = t;
    __syncthreads();
    for (int s = 128; s > 0; s >>= 1) {
        if (threadIdx.x < s) s_red[threadIdx.x] += s_red[threadIdx.x + s];
        __syncthreads();
    }
    if (threadIdx.x == 0) partial[blockIdx.x] = s_red[0];
}

// ---------------------------------------------------------------------------
// Kernel 4: single wave (32 lanes, EXEC all-ones as WMMA requires).
// 8x V_WMMA_F32_16X16X4_F32 with all-ones B reduces the 512 partials in fp32:
// each WMMA sums a 16x4 A-tile (64 partials) into per-row sums; summing the
// 8 accumulator VGPRs per lane and folding across the lane-16 split via
// shfl_xor yields the grand total. Deterministic and exact-fp32.
// ---------------------------------------------------------------------------
__global__ void final_wmma_kernel(const float* __restrict__ partial,
                                  float* __restrict__ out) {
    const int l = threadIdx.x;                  // 0..31, all active
    v8f acc = {};
    v2f bones;
    bones.x = 1.0f;
    bones.y = 1.0f;
#pragma unroll
    for (int i = 0; i < 8; ++i) {
        v2f a;
        a.x = partial[i * 64 + 2 * l];
        a.y = partial[i * 64 + 2 * l + 1];
        // (neg_a, A, neg_b, B, c_mod, C, reuse_a, reuse_b)
        acc = __builtin_amdgcn_wmma_f32_16x16x4_f32(
            false, a, false, bones, (short)0, acc, false, false);
    }
    float s = acc[0] + acc[1] + acc[2] + acc[3] + acc[4] + acc[5] + acc[6] + acc[7];
    s += __shfl_xor(s, 16, 32);                 // fold rows 0-7 with rows 8-15 of column l
    if (l == 0) {
        out[0] = s * (1.0f / (float)(DB * DC * HWSZ));
    }
}

// ---------------------------------------------------------------------------
extern "C" void kernel_launch(void* const* d_in, const int* in_sizes, int n_in,
                              void* d_out, int out_size, void* d_ws, size_t ws_size,
                              hipStream_t stream) {
    (void)in_sizes; (void)n_in; (void)out_size; (void)ws_size;

    const float* x = (const float*)d_in[0];     // [B,C,H,W] f32
    const int*   y = (const int*)d_in[1];       // [B,1,H,W] int
    float* out = (float*)d_out;                 // scalar f32

    char* ws = (char*)d_ws;
    float* field   = (float*)ws;                                  // B*C*H*W f32 (g^2 then d2), 2 MB
    int*   maxd2   = (int*)(ws + (size_t)DB * DC * HWSZ * 4);     // 8 ints (float bits)
    float* partial = (float*)(ws + (size_t)DB * DC * HWSZ * 4 + 128); // 512 f32

    edt_cols_kernel<<<8, 256, 0, stream>>>(y, field, maxd2);            // 2048 columns
    edt_rows_kernel<<<DB * DC * DH, 256, 0, stream>>>(field, maxd2);    // 2048 rows
    loss_partial_kernel<<<(DB * HWSZ) / 256, 256, 0, stream>>>(x, field, maxd2, partial); // 512 blocks
    final_wmma_kernel<<<1, 32, 0, stream>>>(partial, out);
}